// Sparse_attention_25357486916379
// MI455X (gfx1250) — compile-verified
//
#include <hip/hip_runtime.h>
#include <cstdint>

#define T_DIM      4096
#define BLOCK      256
#define PER_THREAD 16      // 4096 / 256
#define TOPK       8
#define EPS_F      1e-7f

typedef float v4f __attribute__((ext_vector_type(4)));

__device__ __forceinline__ float wave_max32(float v) {
#pragma unroll
  for (int o = 16; o > 0; o >>= 1) v = fmaxf(v, __shfl_xor(v, o, 32));
  return v;
}
__device__ __forceinline__ float wave_sum32(float v) {
#pragma unroll
  for (int o = 16; o > 0; o >>= 1) v += __shfl_xor(v, o, 32);
  return v;
}
__device__ __forceinline__ int wave_sum32_i(int v) {
#pragma unroll
  for (int o = 16; o > 0; o >>= 1) v += __shfl_xor(v, o, 32);
  return v;
}

__global__ __launch_bounds__(BLOCK) void sparse_attn_topk_kernel(
    const float* __restrict__ in, float* __restrict__ out, int rows) {
  __shared__ float s_data[T_DIM];   // one full row staged via async copy
  __shared__ float s_redf[8];       // cross-wave reduce scratch (8 waves)
  __shared__ int   s_redi[8];

  const int tid  = threadIdx.x;
  const int lane = tid & 31;
  const int wid  = tid >> 5;
  const int row  = blockIdx.x;
  if (row >= rows) return;

  const float* __restrict__ rowp = in  + (size_t)row * T_DIM;
  float* __restrict__       outp = out + (size_t)row * T_DIM;

  // ---- Stage row -> LDS with CDNA5 async global->LDS b128 copies (ASYNCcnt) ----
  const uint32_t lds_base = (uint32_t)(uintptr_t)(void*)s_data; // low 32 bits = LDS offset
#pragma unroll
  for (int i = 0; i < 4; ++i) {
    uint32_t la = lds_base + (uint32_t)(tid * 64 + i * 16);
    uint64_t ga = (uint64_t)(uintptr_t)(rowp + tid * PER_THREAD + i * 4);
    asm volatile("global_load_async_to_lds_b128 %0, %1, off"
                 :: "v"(la), "v"(ga) : "memory");
  }
  asm volatile("s_wait_asynccnt 0" ::: "memory");
  __syncthreads();

  // ---- Pull my 16 contiguous values into registers (b128 LDS reads) ----
  float v[PER_THREAD];
#pragma unroll
  for (int i = 0; i < 4; ++i) {
    v4f q = reinterpret_cast<const v4f*>(s_data + tid * PER_THREAD)[i];
    v[4 * i + 0] = q.x; v[4 * i + 1] = q.y;
    v[4 * i + 2] = q.z; v[4 * i + 3] = q.w;
  }

  // ---- Branchless local top-8 (sorted descending) ----
  float t[TOPK];
#pragma unroll
  for (int j = 0; j < TOPK; ++j) t[j] = -INFINITY;
#pragma unroll
  for (int i = 0; i < PER_THREAD; ++i) {
    float x = v[i];
#pragma unroll
    for (int j = 0; j < TOPK; ++j) {
      float hi = fmaxf(t[j], x);
      float lo = fminf(t[j], x);
      t[j] = hi;
      x = lo;
    }
  }

  // ---- Block-wide kth-largest (k=8), duplicate-safe, over union of local top-8s ----
  float thresh = INFINITY;
  float kth = -INFINITY;
  int below = 0;
  for (int iter = 0; iter < TOPK; ++iter) {
    float lmax = -INFINITY;
#pragma unroll
    for (int j = 0; j < TOPK; ++j)
      lmax = (t[j] < thresh) ? fmaxf(lmax, t[j]) : lmax;

    lmax = wave_max32(lmax);
    __syncthreads();
    if (lane == 0) s_redf[wid] = lmax;
    __syncthreads();
    float m = s_redf[lane & 7];
#pragma unroll
    for (int o = 4; o > 0; o >>= 1) m = fmaxf(m, __shfl_xor(m, o, 32));

    int c = 0;
#pragma unroll
    for (int j = 0; j < TOPK; ++j) c += (t[j] == m) ? 1 : 0;
    c = wave_sum32_i(c);
    __syncthreads();
    if (lane == 0) s_redi[wid] = c;
    __syncthreads();
    int C = s_redi[lane & 7];
#pragma unroll
    for (int o = 4; o > 0; o >>= 1) C += __shfl_xor(C, o, 32);

    // m, C, below are block-uniform -> convergent break
    if (below + C >= TOPK) { kth = m; break; }
    below += C;
    thresh = m;
  }

  const float delta = kth + EPS_F;

  // ---- Clip + row sum ----
  float w[PER_THREAD];
  float lsum = 0.f;
#pragma unroll
  for (int i = 0; i < PER_THREAD; ++i) {
    float x = fmaxf(v[i] - delta, 0.f);
    w[i] = x;
    lsum += x;
  }
  lsum = wave_sum32(lsum);
  __syncthreads();
  if (lane == 0) s_redf[wid] = lsum;
  __syncthreads();
  float S = s_redf[lane & 7];
#pragma unroll
  for (int o = 4; o > 0; o >>= 1) S += __shfl_xor(S, o, 32);

  const float inv = 1.0f / (S + EPS_F);

  // ---- Streaming (non-temporal) b128 stores of the normalized row ----
  v4f* o4 = reinterpret_cast<v4f*>(outp);
#pragma unroll
  for (int i = 0; i < 4; ++i) {
    v4f q;
    q.x = w[4 * i + 0] * inv;
    q.y = w[4 * i + 1] * inv;
    q.z = w[4 * i + 2] * inv;
    q.w = w[4 * i + 3] * inv;
    __builtin_nontemporal_store(q, o4 + tid * 4 + i);
  }
}

extern "C" void kernel_launch(void* const* d_in, const int* in_sizes, int n_in,
                              void* d_out, int out_size, void* d_ws, size_t ws_size,
                              hipStream_t stream) {
  (void)n_in; (void)out_size; (void)d_ws; (void)ws_size;
  const float* attn = (const float*)d_in[0];
  float* out = (float*)d_out;
  const int rows = in_sizes[0] / T_DIM;   // 16384
  dim3 grid(rows), block(BLOCK);
  sparse_attn_topk_kernel<<<grid, block, 0, stream>>>(attn, out, rows);
}